// GRLVQ_17858474017285
// MI455X (gfx1250) — compile-verified
//
#include <hip/hip_runtime.h>
#include <hip/hip_bf16.h>

typedef __attribute__((ext_vector_type(2))) float v2f;
typedef __attribute__((ext_vector_type(8))) float v8f;

#define BATCH 65536
#define DIM 128
#define NPROTO 512
#define ODIM 8
#define ROWS_PER_BLOCK 64
#define XPITCH 132   // 132 % 64 == 4 -> conflict-free A-fragment loads

// ---------------------------------------------------------------------------
// Prep: W[p][d] = proto[p][d] * r2[d], repacked as float2 pairs:
//   wpack[(d/2)*NPROTO + p] = { W[p][2j], W[p][2j+1] }   (stored as 2 floats)
// halfPsq[p] = 0.5 * sum_d r2[d] * proto[p][d]^2
// ---------------------------------------------------------------------------
__global__ __launch_bounds__(DIM) void grlvq_prep_kernel(
    const float* __restrict__ protos, const float* __restrict__ relevance,
    float* __restrict__ wpack, float* __restrict__ halfPsq)
{
    const int p = blockIdx.x;
    const int d = threadIdx.x;
    const float r  = relevance[d];
    const float r2 = r * r;
    const float v  = protos[p * DIM + d];
    // float2-pair layout, flat float index: (d>>1)*(NPROTO*2) + p*2 + (d&1)
    wpack[(d >> 1) * (NPROTO * 2) + p * 2 + (d & 1)] = v * r2;

    __shared__ float sm[DIM];
    sm[d] = r2 * v * v;
    __syncthreads();
    for (int off = DIM / 2; off > 0; off >>= 1) {
        if (d < off) sm[d] += sm[d + off];
        __syncthreads();
    }
    if (d == 0) halfPsq[p] = 0.5f * sm[0];
}

// ---------------------------------------------------------------------------
// Main: per block, 64 rows of X vs all 512 prototypes.
// score[b,p] = cross[b,p] - 0.5*p_sq[p]  (argmax == reference argmin of dist;
// x_sq[b] is constant in p and dropped). -0.5*p_sq folded into C init.
// ---------------------------------------------------------------------------
__global__ __launch_bounds__(256) void grlvq_main_kernel(
    const float* __restrict__ x, const float* __restrict__ wpack,
    const float* __restrict__ halfPsq, const float* __restrict__ protoOut,
    float* __restrict__ out)
{
    __shared__ __align__(16) float xlds[ROWS_PER_BLOCK * XPITCH];
    __shared__ float redV[ROWS_PER_BLOCK * 2];
    __shared__ int   redI[ROWS_PER_BLOCK * 2];

    const int tid = threadIdx.x;
    const int b0  = blockIdx.x * ROWS_PER_BLOCK;

    // Stage X tile: 64 rows x 128 f32 (32KB) -> LDS, pitch 132. 8 float4/thread.
    #pragma unroll
    for (int i = 0; i < 8; i++) {
        const int idx = tid + i * 256;      // float4 index, 32 per row
        const int row = idx >> 5;
        const int c4  = idx & 31;
        const float4 v = *(const float4*)(x + (size_t)(b0 + row) * DIM + c4 * 4);
        *(float4*)(xlds + row * XPITCH + c4 * 4) = v;
    }
    __syncthreads();

    const int wave    = tid >> 5;
    const int lane    = tid & 31;
    const int ln      = lane & 15;       // column / row-within-tile select
    const int hi      = lane >> 4;       // half-wave: K/M-half select
    const int rowtile = wave >> 1;       // 4 row-tiles of 16 rows
    const int half    = wave & 1;        // column half: 0 -> p[0,256), 1 -> p[256,512)
    const int p_base  = half * 256;

    // C init = -0.5*p_sq[col]; C/D layout: value depends only on N = ln.
    v8f acc[16];
    #pragma unroll
    for (int t = 0; t < 16; t++) {
        const float c = -halfPsq[p_base + t * 16 + ln];
        acc[t] = (v8f){c, c, c, c, c, c, c, c};
    }

    // A fragment (16x4 f32): lane(ln,hi) holds X[row=ln][k0 + 2*hi + {0,1}]
    const float* aptr = xlds + (rowtile * 16 + ln) * XPITCH + 2 * hi;
    // B fragment (4x16 f32): lane(ln,hi) holds W[p=p_base+t*16+ln][k0 + 2*hi + {0,1}]
    //   = wpack float2 element [(k0/2 + hi)*NPROTO + p]
    const v2f* bbase = (const v2f*)wpack + (size_t)hi * NPROTO + p_base + ln;

    for (int k0 = 0; k0 < DIM; k0 += 4) {
        const v2f a = *(const v2f*)(aptr + k0);
        const v2f* bk = bbase + (k0 >> 1) * NPROTO;
        #pragma unroll
        for (int t = 0; t < 16; t++) {
            const v2f b = bk[t * 16];
            acc[t] = __builtin_amdgcn_wmma_f32_16x16x4_f32(
                false, a, false, b, (short)0, acc[t], false, false);
        }
    }

    // Per-lane argmax over 16 tiles; strict '>' keeps smallest p (ties -> first).
    float bestv[8];
    int   besti[8];
    #pragma unroll
    for (int r = 0; r < 8; r++) { bestv[r] = -3.402823466e38f; besti[r] = 0x7fffffff; }
    #pragma unroll
    for (int t = 0; t < 16; t++) {
        const int p = p_base + t * 16 + ln;
        #pragma unroll
        for (int r = 0; r < 8; r++) {
            const float s = acc[t][r];
            if (s > bestv[r]) { bestv[r] = s; besti[r] = p; }
        }
    }

    // Reduce across the 16 lanes of each half (xor masks stay within half).
    #pragma unroll
    for (int r = 0; r < 8; r++) {
        float v = bestv[r];
        int   i = besti[r];
        #pragma unroll
        for (int m = 1; m < 16; m <<= 1) {
            const float ov = __shfl_xor(v, m, 32);
            const int   oi = __shfl_xor(i, m, 32);
            if (ov > v || (ov == v && oi < i)) { v = ov; i = oi; }
        }
        if (ln == 0) {
            // C/D layout: VGPR r holds row M = r + 8*hi
            const int mrow = rowtile * 16 + r + 8 * hi;
            redV[mrow * 2 + half] = v;
            redI[mrow * 2 + half] = i;
        }
    }
    __syncthreads();

    // Combine the two column halves, gather prototype_outputs[idx][0..7].
    if (tid < ROWS_PER_BLOCK) {
        const float v0 = redV[tid * 2 + 0], v1 = redV[tid * 2 + 1];
        const int   i0 = redI[tid * 2 + 0], i1 = redI[tid * 2 + 1];
        const int idx = (v1 > v0 || (v1 == v0 && i1 < i0)) ? i1 : i0;
        const float4* src = (const float4*)(protoOut + (size_t)idx * ODIM);
        float4* dst = (float4*)(out + (size_t)(b0 + tid) * ODIM);
        dst[0] = src[0];
        dst[1] = src[1];
    }
}

extern "C" void kernel_launch(void* const* d_in, const int* in_sizes, int n_in,
                              void* d_out, int out_size, void* d_ws, size_t ws_size,
                              hipStream_t stream) {
    const float* x         = (const float*)d_in[0];   // [65536,128]
    const float* protos    = (const float*)d_in[1];   // [512,128]
    const float* protoOut  = (const float*)d_in[2];   // [512,8]
    const float* relevance = (const float*)d_in[3];   // [128]
    float* out = (float*)d_out;                       // [65536,8]

    float* wpack   = (float*)d_ws;                    // 64*512*2 = 65536 floats (256 KB)
    float* halfPsq = wpack + (DIM / 2) * NPROTO * 2;  // 512 floats

    grlvq_prep_kernel<<<NPROTO, DIM, 0, stream>>>(protos, relevance, wpack, halfPsq);
    grlvq_main_kernel<<<BATCH / ROWS_PER_BLOCK, 256, 0, stream>>>(
        x, wpack, halfPsq, protoOut, out);
}